// RayTransformer_51496657879764
// MI455X (gfx1250) — compile-verified
//
#include <hip/hip_runtime.h>
#include <hip/hip_bf16.h>

typedef __attribute__((ext_vector_type(2))) float v2f;
typedef __attribute__((ext_vector_type(8))) float v8f;

#define RNN   2048
#define SNN   64
#define NPTS  (RNN * SNN)          // 131072
#define DD    40
#define DP    48
#define D2    80
#define HW    65536                 // 256*256

// ---------------------------------------------------------------------------
// WMMA helper: one 16x16 f32 output tile of  OUT = A[64xKTOT] * B[KROWSxNCOLS]
//   A: LDS, row stride LDA, zero-padded (no bound checks)
//   B: global weights, row stride LDB; branchless zero-padding via clamped
//      index + 0/1 mask so no exec-mask branches appear in the WMMA loop.
// All dims compile-time so masks fold away where possible and the K loop
// fully unrolls. Layouts per CDNA5 ISA 7.12.2 (32-bit A 16x4 / C 16x16).
// ---------------------------------------------------------------------------
template <int LDA, int LDB, int KROWS, int NCOLS, int KTOT>
__device__ __forceinline__ v8f gemm_tile(const float* __restrict__ As,
                                         const float* __restrict__ Bg,
                                         int m0, int n0, int lane) {
    const int lr  = lane & 15;
    const int hi2 = (lane >> 4) << 1;        // 0 or 2
    v8f acc = {};
    const float* arow = As + (m0 + lr) * LDA;
    const int   n     = n0 + lr;
    const int   nc    = (n < NCOLS) ? n : (NCOLS - 1);
    const float nmask = (n < NCOLS) ? 1.0f : 0.0f;
#pragma unroll
    for (int k0 = 0; k0 < KTOT; k0 += 4) {
        const int ka = k0 + hi2;
        // compile-time foldable row masks (KROWS is a multiple of 4 here,
        // so ka<KROWS and ka+1<KROWS are uniform per k0 step)
        const float m0f = (ka     < KROWS) ? nmask : 0.0f;
        const float m1f = (ka + 1 < KROWS) ? nmask : 0.0f;
        const int ka0 = (ka     < KROWS) ? ka     : (KROWS - 1);
        const int ka1 = (ka + 1 < KROWS) ? ka + 1 : (KROWS - 1);
        v2f a, b;
        a.x = arow[ka];
        a.y = arow[ka + 1];
        b.x = Bg[ka0 * LDB + nc] * m0f;
        b.y = Bg[ka1 * LDB + nc] * m1f;
        acc = __builtin_amdgcn_wmma_f32_16x16x4_f32(false, a, false, b,
                                                    (short)0, acc, false, false);
    }
    return acc;
}

__device__ __forceinline__ void store_tile(float* __restrict__ Cs, int ldc,
                                           int m0, int n0, int lane, v8f acc,
                                           bool do_relu) {
    const int lr = lane & 15;
    const int mh = (lane >> 4) * 8;
#pragma unroll
    for (int i = 0; i < 8; ++i) {
        float v = acc[i];
        if (do_relu) v = fmaxf(v, 0.0f);
        Cs[(m0 + i + mh) * ldc + n0 + lr] = v;
    }
}

// ---------------------------------------------------------------------------
// Kernel A: projection, bilinear sampling, radiance MLP + softmax blend,
//           and x = [feat_view0 | posenc] staged into the ray_feat region.
// One thread per point; gathers are L2-resident (42MB feat << 192MB L2).
// ---------------------------------------------------------------------------
__global__ __launch_bounds__(256) void sample_radiance_kernel(
    const float* __restrict__ point3D,     // [NPTS,3]
    const float* __restrict__ src_imgs,    // [5,3,256,256]
    const float* __restrict__ src_feat,    // [5,32,256,256]
    const float* __restrict__ poses,       // [5,4,4]
    const float* __restrict__ poses_inv,   // [5,4,4]
    const float* __restrict__ ref_inv,     // [4,4]
    const float* __restrict__ rw1, const float* __restrict__ rb1,
    const float* __restrict__ rw2, const float* __restrict__ rb2,
    const float* __restrict__ rw3, const float* __restrict__ rb3,
    float* __restrict__ out_rad,           // [NPTS,3]
    float* __restrict__ out_pip,           // [5,2,NPTS]
    float* __restrict__ out_x)             // [NPTS,40]  (aliases ray_feat out)
{
    __shared__ float s_w1[35 * 16];
    __shared__ float s_b1[16];
    __shared__ float s_w2[16 * 8];
    __shared__ float s_b2[8];
    __shared__ float s_w3[8];
    __shared__ float s_b3;

    for (int i = threadIdx.x; i < 35 * 16; i += 256) s_w1[i] = rw1[i];
    for (int i = threadIdx.x; i < 128; i += 256)     s_w2[i] = rw2[i];
    if (threadIdx.x < 16) s_b1[threadIdx.x] = rb1[threadIdx.x];
    if (threadIdx.x < 8) { s_b2[threadIdx.x] = rb2[threadIdx.x];
                           s_w3[threadIdx.x] = rw3[threadIdx.x]; }
    if (threadIdx.x == 0) s_b3 = rb3[0];
    __syncthreads();

    const int p = blockIdx.x * 256 + threadIdx.x;   // < NPTS
    const int s = p & (SNN - 1);

    const float px = point3D[p * 3 + 0];
    const float py = point3D[p * 3 + 1];
    const float pz = point3D[p * 3 + 2];

    // v1 = normalize(pt - ref translation)
    float rx = px - ref_inv[3], ry = py - ref_inv[7], rz = pz - ref_inv[11];
    float rn = rsqrtf(rx * rx + ry * ry + rz * rz);
    const float v1x = rx * rn, v1y = ry * rn, v1z = rz * rn;

    float logit[4];
    float rgbv[4][3];

    for (int v = 0; v < 5; ++v) {
        const float* P = poses + v * 16;
        const float hx = P[0] * px + P[1] * py + P[2]  * pz + P[3];
        const float hy = P[4] * px + P[5] * py + P[6]  * pz + P[7];
        const float hz = P[8] * px + P[9] * py + P[10] * pz + P[11];
        const float gx = hx / hz;
        const float gy = hy / hz;
        out_pip[(v * 2 + 0) * NPTS + p] = gx;
        out_pip[(v * 2 + 1) * NPTS + p] = gy;

        const float dok = (hz > 0.0f) ? 1.0f : 0.0f;
        const float inb = (gx >= 0.0f && gx <= 255.0f &&
                           gy >= 0.0f && gy <= 255.0f) ? 1.0f : 0.0f;

        const float x0f = floorf(gx), y0f = floorf(gy);
        const float wx = gx - x0f, wy = gy - y0f;
        const float x1f = x0f + 1.0f, y1f = y0f + 1.0f;

        const float okx0 = (x0f >= 0.0f && x0f <= 255.0f) ? 1.0f : 0.0f;
        const float okx1 = (x1f >= 0.0f && x1f <= 255.0f) ? 1.0f : 0.0f;
        const float oky0 = (y0f >= 0.0f && y0f <= 255.0f) ? 1.0f : 0.0f;
        const float oky1 = (y1f >= 0.0f && y1f <= 255.0f) ? 1.0f : 0.0f;

        const float w00 = (1.0f - wx) * (1.0f - wy) * okx0 * oky0;
        const float w10 = wx * (1.0f - wy) * okx1 * oky0;
        const float w01 = (1.0f - wx) * wy * okx0 * oky1;
        const float w11 = wx * wy * okx1 * oky1;

        const int ix0 = min(max((int)x0f, 0), 255);
        const int ix1 = min(max((int)x0f + 1, 0), 255);
        const int iy0 = min(max((int)y0f, 0), 255);
        const int iy1 = min(max((int)y0f + 1, 0), 255);
        const int o00 = iy0 * 256 + ix0;
        const int o10 = iy0 * 256 + ix1;
        const int o01 = iy1 * 256 + ix0;
        const int o11 = iy1 * 256 + ix1;

        if (v == 0) {
            const float* F = src_feat;
            for (int c = 0; c < 32; ++c) {
                const float* Fc = F + c * HW;
                out_x[p * DD + c] = w00 * Fc[o00] + w10 * Fc[o10] +
                                    w01 * Fc[o01] + w11 * Fc[o11];
            }
            // order_posenc(8, 64): scales 1, 0.1, 0.01, 0.001
            const float sf = (float)s;
            out_x[p * DD + 32] = __sinf(sf);
            out_x[p * DD + 33] = __cosf(sf);
            out_x[p * DD + 34] = __sinf(sf * 0.1f);
            out_x[p * DD + 35] = __cosf(sf * 0.1f);
            out_x[p * DD + 36] = __sinf(sf * 0.01f);
            out_x[p * DD + 37] = __cosf(sf * 0.01f);
            out_x[p * DD + 38] = __sinf(sf * 0.001f);
            out_x[p * DD + 39] = __cosf(sf * 0.001f);
        } else {
            const int vi = v - 1;
            const float* F = src_feat + v * 32 * HW;
            float h1[16];
#pragma unroll
            for (int j = 0; j < 16; ++j) h1[j] = s_b1[j];
            for (int c = 0; c < 32; ++c) {
                const float* Fc = F + c * HW;
                const float f = w00 * Fc[o00] + w10 * Fc[o10] +
                                w01 * Fc[o01] + w11 * Fc[o11];
#pragma unroll
                for (int j = 0; j < 16; ++j) h1[j] += f * s_w1[c * 16 + j];
            }
            // relative viewing direction
            const float* PI = poses_inv + v * 16;
            float dx = px - PI[3], dy = py - PI[7], dz = pz - PI[11];
            float dn = rsqrtf(dx * dx + dy * dy + dz * dz);
            const float d0 = v1x - dx * dn;
            const float d1 = v1y - dy * dn;
            const float d2 = v1z - dz * dn;
#pragma unroll
            for (int j = 0; j < 16; ++j)
                h1[j] += d0 * s_w1[32 * 16 + j] + d1 * s_w1[33 * 16 + j] +
                         d2 * s_w1[34 * 16 + j];
            float h2[8];
#pragma unroll
            for (int j = 0; j < 8; ++j) {
                float a = s_b2[j];
#pragma unroll
                for (int i2 = 0; i2 < 16; ++i2)
                    a += fmaxf(h1[i2], 0.0f) * s_w2[i2 * 8 + j];
                h2[j] = fmaxf(a, 0.0f);
            }
            float lg = s_b3;
#pragma unroll
            for (int j = 0; j < 8; ++j) lg += h2[j] * s_w3[j];
            const float m = inb * dok;
            logit[vi] = (m == 0.0f) ? -1000000000.0f : lg;

            const float* R = src_imgs + v * 3 * HW;
#pragma unroll
            for (int c = 0; c < 3; ++c) {
                const float* Rc = R + c * HW;
                rgbv[vi][c] = w00 * Rc[o00] + w10 * Rc[o10] +
                              w01 * Rc[o01] + w11 * Rc[o11];
            }
        }
    }

    // softmax over the 4 source views, blend rgb
    const float mx = fmaxf(fmaxf(logit[0], logit[1]), fmaxf(logit[2], logit[3]));
    const float e0 = __expf(logit[0] - mx);
    const float e1 = __expf(logit[1] - mx);
    const float e2 = __expf(logit[2] - mx);
    const float e3 = __expf(logit[3] - mx);
    const float inv = 1.0f / (e0 + e1 + e2 + e3);
#pragma unroll
    for (int c = 0; c < 3; ++c)
        out_rad[p * 3 + c] = (e0 * rgbv[0][c] + e1 * rgbv[1][c] +
                              e2 * rgbv[2][c] + e3 * rgbv[3][c]) * inv;
}

// ---------------------------------------------------------------------------
// Kernel B: LoFTR self-attention layer, one workgroup (4 waves) per ray.
// All six GEMM stages use V_WMMA_F32_16X16X4_F32.
// ---------------------------------------------------------------------------
__global__ __launch_bounds__(128) void rayformer_kernel(
    const float* __restrict__ xg,     // [NPTS,40]  (== ray_feat region)
    const float* __restrict__ wq, const float* __restrict__ wk,
    const float* __restrict__ wv, const float* __restrict__ wm,
    const float* __restrict__ ln1g, const float* __restrict__ ln1b,
    const float* __restrict__ w1, const float* __restrict__ w2,
    const float* __restrict__ ln2g, const float* __restrict__ ln2b,
    float* __restrict__ outg)         // [NPTS,40]
{
    __shared__ float Xs[SNN * DP];        // 3072
    __shared__ float S[4 * SNN * DP];     // 12288 recycled scratch
    __shared__ float KVs[DD * 5];
    __shared__ float Ksm[DD];

    const int r    = blockIdx.x;
    const int tid  = threadIdx.x;
    const int wave = tid >> 5;
    const int lane = tid & 31;

    // stage x (zero-padded 40 -> 48)
    for (int i = tid; i < SNN * DP; i += 128) {
        const int row = i / DP, col = i % DP;
        Xs[i] = (col < DD) ? xg[(r * SNN + row) * DD + col] : 0.0f;
    }
    __syncthreads();

    float* Qs = S;
    float* Ks = S + SNN * DP;
    float* Vs = S + 2 * SNN * DP;
    float* Ms = S + 3 * SNN * DP;

    // q,k,v projections: 3 x (4x3) tiles
    for (int j = wave; j < 36; j += 4) {
        const int which = j / 12, t = j % 12;
        const int m0 = (t / 3) * 16, n0 = (t % 3) * 16;
        const float* W = (which == 0) ? wq : ((which == 1) ? wk : wv);
        float* O = (which == 0) ? Qs : ((which == 1) ? Ks : Vs);
        v8f acc = gemm_tile<DP, DD, DD, DD, DP>(Xs, W, m0, n0, lane);
        store_tile(O, DP, m0, n0, lane, acc, false);
    }
    __syncthreads();

    // elu(x)+1 on Q and K.  Padded cols are transformed too (0 -> 1) which is
    // harmless: attention reads only cols < 40 and the Q/K regions are
    // recycled (fully rewritten) afterwards.  No div/mod needed.
    for (int i = tid; i < SNN * DP; i += 128) {
        const float q = Qs[i];
        Qs[i] = (q > 0.0f) ? q + 1.0f : __expf(q);
        const float k = Ks[i];
        Ks[i] = (k > 0.0f) ? k + 1.0f : __expf(k);
    }
    __syncthreads();

    // linear attention stage 1: Ksum[d], KV[d][e] per (head,dim)
    if (tid < DD) {
        const int h = tid / 5;
        float ks = 0.0f;
        float kv[5] = {0, 0, 0, 0, 0};
        for (int ss = 0; ss < SNN; ++ss) {
            const float kk = Ks[ss * DP + tid];
            ks += kk;
#pragma unroll
            for (int e = 0; e < 5; ++e)
                kv[e] += kk * Vs[ss * DP + h * 5 + e] * (1.0f / SNN);
        }
        Ksm[tid] = ks;
#pragma unroll
        for (int e = 0; e < 5; ++e) KVs[tid * 5 + e] = kv[e];
    }
    __syncthreads();

    // stage 2: per-sample output
    if (tid < SNN) {
        const int l = tid;
#pragma unroll
        for (int h = 0; h < 8; ++h) {
            float z = 0.0f;
#pragma unroll
            for (int d = 0; d < 5; ++d) z += Qs[l * DP + h * 5 + d] * Ksm[h * 5 + d];
            const float Z = 1.0f / (z + 1e-6f);
#pragma unroll
            for (int e = 0; e < 5; ++e) {
                float o = 0.0f;
#pragma unroll
                for (int d = 0; d < 5; ++d)
                    o += Qs[l * DP + h * 5 + d] * KVs[(h * 5 + d) * 5 + e];
                Ms[l * DP + h * 5 + e] = o * Z * (float)SNN;
            }
        }
#pragma unroll
        for (int c = DD; c < DP; ++c) Ms[l * DP + c] = 0.0f;
    }
    __syncthreads();

    // msg @ wm  -> MM (reuse V region)
    float* MM = Vs;
    for (int t = wave; t < 12; t += 4) {
        const int m0 = (t / 3) * 16, n0 = (t % 3) * 16;
        v8f acc = gemm_tile<DP, DD, DD, DD, DP>(Ms, wm, m0, n0, lane);
        store_tile(MM, DP, m0, n0, lane, acc, false);
    }
    __syncthreads();

    // layernorm1 + build Cat = [x | ln(msg)]  (Cat at S[0..5120), Q/K dead)
    float* Cat = S;
    if (tid < SNN) {
        const int row = tid;
        float mu = 0.0f;
        for (int c = 0; c < DD; ++c) mu += MM[row * DP + c];
        mu *= (1.0f / DD);
        float var = 0.0f;
        for (int c = 0; c < DD; ++c) {
            const float d = MM[row * DP + c] - mu;
            var += d * d;
        }
        const float is = rsqrtf(var * (1.0f / DD) + 1e-5f);
        for (int c = 0; c < DD; ++c) {
            Cat[row * D2 + c] = Xs[row * DP + c];
            Cat[row * D2 + DD + c] =
                (MM[row * DP + c] - mu) * is * ln1g[c] + ln1b[c];
        }
    }
    __syncthreads();

    // H1 = relu(Cat @ mlp_w1)   (H1 at S+6144, MM dead; no masking at all)
    float* H1 = S + 2 * SNN * DP;
    for (int t = wave; t < 20; t += 4) {
        const int m0 = (t / 5) * 16, n0 = (t % 5) * 16;
        v8f acc = gemm_tile<D2, D2, D2, D2, D2>(Cat, w1, m0, n0, lane);
        store_tile(H1, D2, m0, n0, lane, acc, true);
    }
    __syncthreads();

    // O2 = H1 @ mlp_w2          (O2 at S[0..3072), Cat dead)
    float* O2 = S;
    for (int t = wave; t < 12; t += 4) {
        const int m0 = (t / 3) * 16, n0 = (t % 3) * 16;
        v8f acc = gemm_tile<D2, DD, D2, DD, D2>(H1, w2, m0, n0, lane);
        store_tile(O2, DP, m0, n0, lane, acc, false);
    }
    __syncthreads();

    // layernorm2 + residual -> output
    if (tid < SNN) {
        const int row = tid;
        float mu = 0.0f;
        for (int c = 0; c < DD; ++c) mu += O2[row * DP + c];
        mu *= (1.0f / DD);
        float var = 0.0f;
        for (int c = 0; c < DD; ++c) {
            const float d = O2[row * DP + c] - mu;
            var += d * d;
        }
        const float is = rsqrtf(var * (1.0f / DD) + 1e-5f);
        for (int c = 0; c < DD; ++c)
            outg[(r * SNN + row) * DD + c] =
                Xs[row * DP + c] + (O2[row * DP + c] - mu) * is * ln2g[c] + ln2b[c];
    }
}

// ---------------------------------------------------------------------------
extern "C" void kernel_launch(void* const* d_in, const int* in_sizes, int n_in,
                              void* d_out, int out_size, void* d_ws, size_t ws_size,
                              hipStream_t stream) {
    (void)in_sizes; (void)d_ws; (void)ws_size; (void)out_size;
    if (n_in < 22) return;
    const float* point3D   = (const float*)d_in[0];
    const float* src_imgs  = (const float*)d_in[1];
    const float* src_feat  = (const float*)d_in[2];
    const float* poses     = (const float*)d_in[3];
    const float* poses_inv = (const float*)d_in[4];
    const float* ref_inv   = (const float*)d_in[5];
    const float* rw1 = (const float*)d_in[6];
    const float* rb1 = (const float*)d_in[7];
    const float* rw2 = (const float*)d_in[8];
    const float* rb2 = (const float*)d_in[9];
    const float* rw3 = (const float*)d_in[10];
    const float* rb3 = (const float*)d_in[11];
    const float* wq  = (const float*)d_in[12];
    const float* wk  = (const float*)d_in[13];
    const float* wv  = (const float*)d_in[14];
    const float* wm  = (const float*)d_in[15];
    const float* ln1g = (const float*)d_in[16];
    const float* ln1b = (const float*)d_in[17];
    const float* w1   = (const float*)d_in[18];
    const float* w2   = (const float*)d_in[19];
    const float* ln2g = (const float*)d_in[20];
    const float* ln2b = (const float*)d_in[21];

    float* out = (float*)d_out;
    float* out_rad = out;                        // NPTS*3   = 393216
    float* out_pip = out + NPTS * 3;             // 5*2*NPTS = 1310720
    float* out_rf  = out + NPTS * 3 + 10 * NPTS; // NPTS*40  = 5242880

    // Kernel A: sampling + radiance; writes x into the ray_feat region.
    sample_radiance_kernel<<<NPTS / 256, 256, 0, stream>>>(
        point3D, src_imgs, src_feat, poses, poses_inv, ref_inv,
        rw1, rb1, rw2, rb2, rw3, rb3,
        out_rad, out_pip, out_rf);

    // Kernel B: transformer layer; reads x from (and overwrites) ray_feat.
    rayformer_kernel<<<RNN, 128, 0, stream>>>(
        out_rf, wq, wk, wv, wm, ln1g, ln1b, w1, w2, ln2g, ln2b, out_rf);
}